// BinarizeLayer_61546881352475
// MI455X (gfx1250) — compile-verified
//
#include <hip/hip_runtime.h>
#include <cstdint>

// Graph-cut ICM binarization for MI455X (gfx1250).
//
// Algebra: with lambda=0.5 the ICM update (cost1 < cost0) reduces to
//     label_new = ( nbr_sum > T ),  T = log((1-p+eps)/(p+eps)) + 0.5*ncnt
// T is precomputed once (into parity-split planes for dense reads) and each
// of the 60 half-sweeps is a pure 4-neighbor stencil on 0/1 float labels.
//
// Roofline: T (64MB, in d_ws) + labels (64MB, in d_out) = 128MB < 192MB L2,
// so all 60 half-sweeps run out of L2 (~167MB touched each); only init
// streams HBM (~192MB ~= 8us at 23.3 TB/s). Compute is ~25 MFLOP/sweep ->
// L2-bandwidth + launch bound. CDNA5 paths: per-lane async global->LDS DMA
// (b128 body + b32 halo, ASYNCcnt / s_wait_asynccnt) + global_prefetch of T.

#define IMG_H 4096
#define IMG_W 4096
#define HALF_W (IMG_W / 2)
#define PLANE ((size_t)IMG_H * HALF_W)   // one parity plane of T (floats)
#define EPSF  1e-6f
#define TW    128                // tile width
#define TH    64                 // tile height
#define LW    136                // LDS row stride: halo@3, body@4..131, halo@132
#define N_HALF_SWEEPS 60         // 30 sweeps x 2 parities

#if defined(__AMDGCN__) && \
    __has_builtin(__builtin_amdgcn_global_load_async_to_lds_b32) && \
    __has_builtin(__builtin_amdgcn_s_wait_asynccnt)
#define HAVE_ASYNC 1
#else
#define HAVE_ASYNC 0
#endif

#if defined(__AMDGCN__) && \
    __has_builtin(__builtin_amdgcn_global_load_async_to_lds_b128) && HAVE_ASYNC
#define HAVE_ASYNC128 1
#else
#define HAVE_ASYNC128 0
#endif

#if HAVE_ASYNC
typedef __attribute__((address_space(1))) int gas_i32;   // global scalar
typedef __attribute__((address_space(3))) int las_i32;   // LDS scalar
typedef int v4i __attribute__((ext_vector_type(4)));
typedef __attribute__((address_space(1))) v4i gas_v4i;   // global 16B vector
typedef __attribute__((address_space(3))) v4i las_v4i;   // LDS 16B vector
#endif

__device__ __forceinline__ void lds_copy_f32(const float* g, float* l) {
#if HAVE_ASYNC
  __builtin_amdgcn_global_load_async_to_lds_b32(
      (gas_i32*)(uintptr_t)g, (las_i32*)(unsigned int)(uintptr_t)l, 0, 0);
#else
  *l = *g;
#endif
}

__device__ __forceinline__ void lds_copy_f32x4(const float* g, float* l) {
#if HAVE_ASYNC128
  __builtin_amdgcn_global_load_async_to_lds_b128(
      (gas_v4i*)(uintptr_t)g, (las_v4i*)(unsigned int)(uintptr_t)l, 0, 0);
#elif HAVE_ASYNC
#pragma unroll
  for (int e = 0; e < 4; ++e)
    __builtin_amdgcn_global_load_async_to_lds_b32(
        (gas_i32*)(uintptr_t)(g + e), (las_i32*)(unsigned int)(uintptr_t)(l + e), 0, 0);
#else
  *(float4*)l = *(const float4*)g;
#endif
}

__device__ __forceinline__ void lds_copy_wait() {
#if HAVE_ASYNC
  __builtin_amdgcn_s_wait_asynccnt(0);
#endif
}

// ---------------------------------------------------------------------------
// Init: T -> parity-split planes in ws; L0 = (p > 0.5) interleaved in d_out.
// ---------------------------------------------------------------------------
__global__ __launch_bounds__(256) void binarize_init(
    const float* __restrict__ probs, float* __restrict__ thr,
    float* __restrict__ lab) {
  int idx = (blockIdx.x * 256 + threadIdx.x) * 4;
  int i = idx >> 12;               // / IMG_W
  int jb = idx & (IMG_W - 1);      // % IMG_W, multiple of 4

  float4 p4 = *(const float4*)(probs + idx);
  float pv[4] = {p4.x, p4.y, p4.z, p4.w};
  float tv[4], lv[4];
  float nci = (float)((i > 0) + (i < IMG_H - 1));
#pragma unroll
  for (int e = 0; e < 4; ++e) {
    int j = jb + e;
    float nc = nci + (float)((j > 0) + (j < IMG_W - 1));
    float p = pv[e];
    tv[e] = logf((1.0f - p + EPSF) / (p + EPSF)) + 0.5f * nc;
    lv[e] = (p > 0.5f) ? 1.0f : 0.0f;
  }
  // parity of (i+j): jb%4==0 so elements e=0..3 alternate starting at (i&1)
  int pr = i & 1;
  size_t rowoff = (size_t)i * HALF_W + (jb >> 1);
  *(float2*)(thr + (size_t)pr * PLANE + rowoff)       = make_float2(tv[0], tv[2]);
  *(float2*)(thr + (size_t)(pr ^ 1) * PLANE + rowoff) = make_float2(tv[1], tv[3]);
  *(float4*)(lab + idx) = make_float4(lv[0], lv[1], lv[2], lv[3]);
}

// ---------------------------------------------------------------------------
// One checkerboard half-sweep. Block = TWxTH tile; tile + 1-cell halo of L is
// async-DMA'd into LDS (out-of-image cells stay 0 == reference zero pad).
// Only opposite-parity LDS entries feed ns, only `parity` cells change; the
// partner cell in each float2 store is rewritten with its identical old
// value, so concurrent blocks never race on bytes that differ.
// ---------------------------------------------------------------------------
__global__ __launch_bounds__(256) void binarize_sweep(
    const float* __restrict__ thrp, float* __restrict__ lab, int parity) {
  __shared__ float ls[(TH + 2) * LW];
  const int i0 = blockIdx.y * TH;
  const int j0 = blockIdx.x * TW;
  const int tid = threadIdx.x;
  const bool border =
      (i0 == 0) | (j0 == 0) | (i0 + TH == IMG_H) | (j0 + TW == IMG_W);

  // 1) zero-pad halo needed only for image-border tiles
  if (border) {
    for (int k = tid; k < (TH + 2) * LW; k += 256) ls[k] = 0.0f;
    __syncthreads();
  }

  // 2a) async-stage body: rows i0-1..i0+TH, cols j0..j0+127 as b128 chunks
  for (int k = tid; k < (TH + 2) * 32; k += 256) {
    int r = k >> 5;                    // 0..65
    int q = k & 31;                    // 16B chunk in row
    int gi = i0 - 1 + r;
    if ((unsigned)gi < (unsigned)IMG_H)
      lds_copy_f32x4(lab + (size_t)gi * IMG_W + j0 + 4 * q,
                     &ls[r * LW + 4 + 4 * q]);
  }
  // 2b) async-stage halo columns (j0-1 and j0+TW) as b32
  for (int k = tid; k < (TH + 2) * 2; k += 256) {
    int r = k >> 1;
    int side = k & 1;
    int gi = i0 - 1 + r;
    int gj = side ? (j0 + TW) : (j0 - 1);
    if ((unsigned)gi < (unsigned)IMG_H && (unsigned)gj < (unsigned)IMG_W)
      lds_copy_f32(lab + (size_t)gi * IMG_W + gj,
                   &ls[r * LW + (side ? (TW + 4) : 3)]);
  }

  // 3) overlap: prefetch this tile's dense T-plane lines while DMA runs
  const float* tp = thrp + (size_t)parity * PLANE + (size_t)i0 * HALF_W + (j0 >> 1);
  {
    int r = tid >> 2;                  // 64 rows x 4 cachelines (64B) each
    int seg = tid & 3;
    __builtin_prefetch(tp + (size_t)r * HALF_W + seg * 16, 0, 3);
  }

  lds_copy_wait();       // this wave's ASYNCcnt -> 0
  __syncthreads();       // all waves' tile data resident

  // 4) update `parity` cells: 64 rows x 64 cells, 16 per thread
  for (int k = tid; k < TH * (TW / 2); k += 256) {
    int r = k >> 6;                    // tile row
    int m = k & 63;                    // cell pair index
    int bit = (r + parity) & 1;        // i0,j0 even => global parity check
    int C = 2 * m + bit + 4;           // LDS col of updated cell
    float ns = ls[r * LW + C]                  // up    (LDS row r   = i-1)
             + ls[(r + 2) * LW + C]            // down  (LDS row r+2 = i+1)
             + ls[(r + 1) * LW + C - 1]        // left
             + ls[(r + 1) * LW + C + 1];       // right
    float tval = tp[(size_t)r * HALF_W + m];
    float nv = (ns > tval) ? 1.0f : 0.0f;
    float pvv = ls[(r + 1) * LW + 2 * m + (bit ^ 1) + 4];  // partner (unchanged)
    float2 st = bit ? make_float2(pvv, nv) : make_float2(nv, pvv);
    *(float2*)(lab + (size_t)(i0 + r) * IMG_W + j0 + 2 * m) = st;
  }
}

// ---------------------------------------------------------------------------
extern "C" void kernel_launch(void* const* d_in, const int* in_sizes, int n_in,
                              void* d_out, int out_size, void* d_ws,
                              size_t ws_size, hipStream_t stream) {
  (void)in_sizes; (void)n_in; (void)out_size; (void)ws_size;
  const float* probs = (const float*)d_in[0];
  float* lab = (float*)d_out;          // labels live in the output buffer
  float* thr = (float*)d_ws;           // 64MB parity-split threshold planes

  dim3 ib(256), ig((IMG_H * IMG_W / 4) / 256);
  binarize_init<<<ig, ib, 0, stream>>>(probs, thr, lab);

  dim3 sb(256), sg(IMG_W / TW, IMG_H / TH);
  for (int s = 0; s < N_HALF_SWEEPS; ++s)
    binarize_sweep<<<sg, sb, 0, stream>>>(thr, lab, s & 1);
}